// GCN_FAN_65163243815243
// MI455X (gfx1250) — compile-verified
//
#include <hip/hip_runtime.h>

typedef __attribute__((ext_vector_type(2))) float v2f;
typedef __attribute__((ext_vector_type(8))) float v8f;

#define NEG_SLOPE 0.01f

// ---------------------------------------------------------------------------
// Generic fp32 WMMA GEMM: C[M,N] = act(A[M,K] @ B[K,N] + bias[N])
// One 16x16 tile per wave, 8 waves per block. Requires M%128==0 handled by
// grid.y = M/128, N%16==0, K%4==0. Uses V_WMMA_F32_16X16X4_F32.
// A layout per lane: lanes0-15 K={k0,k0+1}, lanes16-31 K={k0+2,k0+3} (row=lane%16)
// B layout per lane: same K split, col=lane%16
// D layout: VGPR r -> M = 16*tileM + r + 8*(lane>=16), N = 16*tileN + lane%16
// ---------------------------------------------------------------------------
__global__ __launch_bounds__(256) void wmma_gemm_f32(
    const float* __restrict__ A, const float* __restrict__ B,
    const float* __restrict__ bias, float* __restrict__ C,
    int M, int Nn, int K, int act)
{
    const int lane  = threadIdx.x & 31;
    const int wave  = threadIdx.x >> 5;
    const int half  = lane >> 4;      // 0: K lo pair, 1: K hi pair
    const int l16   = lane & 15;
    const int tileM = blockIdx.y * 8 + wave;
    const int tileN = blockIdx.x;
    const int row   = tileM * 16 + l16;
    const int col   = tileN * 16 + l16;

    v8f acc = {};
    for (int k0 = 0; k0 < K; k0 += 4) {
        const int kk = k0 + half * 2;
        v2f a = *(const v2f*)(A + (size_t)row * K + kk);   // 8B aligned (kk even)
        v2f b;
        b.x = B[(size_t)(kk + 0) * Nn + col];
        b.y = B[(size_t)(kk + 1) * Nn + col];
        acc = __builtin_amdgcn_wmma_f32_16x16x4_f32(
            /*neg_a=*/false, a, /*neg_b=*/false, b,
            /*c_mod=*/(short)0, acc, /*reuse_a=*/false, /*reuse_b=*/false);
    }

    const float bv = bias ? bias[col] : 0.0f;
#pragma unroll
    for (int r = 0; r < 8; ++r) {
        const int m = tileM * 16 + r + 8 * half;
        float v = acc[r] + bv;
        if (act) v = (v >= 0.0f) ? v : NEG_SLOPE * v;
        C[(size_t)m * Nn + col] = v;
    }
}

// ---------------------------------------------------------------------------
// Zero a float buffer (graph-capture-safe alternative to hipMemsetAsync)
// ---------------------------------------------------------------------------
__global__ void zero_kernel(float* __restrict__ p, size_t n)
{
    size_t i = (size_t)blockIdx.x * blockDim.x + threadIdx.x;
    const size_t stride = (size_t)gridDim.x * blockDim.x;
    for (; i < n; i += stride) p[i] = 0.0f;
}

// ---------------------------------------------------------------------------
// deg[dst] += 1 over all E edges (self-loops folded into dinv kernel)
// ---------------------------------------------------------------------------
__global__ void deg_kernel(const long long* __restrict__ ei, float* __restrict__ deg,
                           long long E)
{
    long long i = (long long)blockIdx.x * blockDim.x + threadIdx.x;
    const long long stride = (long long)gridDim.x * blockDim.x;
    for (; i < E; i += stride) {
        const long long d = ei[E + i];           // edge_index[1][i]
        atomicAdd(&deg[d], 1.0f);
    }
}

__global__ void dinv_kernel(const float* __restrict__ deg, float* __restrict__ dinv,
                            int n)
{
    const int i = blockIdx.x * blockDim.x + threadIdx.x;
    if (i < n) dinv[i] = rsqrtf(deg[i] + 1.0f);  // +1 = self loop; always > 0
}

// ---------------------------------------------------------------------------
// Edge scatter: one wave per edge; lane q handles features [4q, 4q+4).
// agg[dst, :] += h[src, :] * dinv[src] * dinv[dst]
// ---------------------------------------------------------------------------
__global__ __launch_bounds__(256) void edge_scatter(
    const float* __restrict__ h, const long long* __restrict__ ei,
    const float* __restrict__ dinv, float* __restrict__ agg, long long E)
{
    const long long tid     = (long long)blockIdx.x * blockDim.x + threadIdx.x;
    const long long nthread = (long long)gridDim.x * blockDim.x;
    const int q = threadIdx.x & 31;
    for (long long g = tid; ; g += nthread) {
        const long long e = g >> 5;              // wave-uniform (nthread % 32 == 0)
        if (e >= E) break;
        const long long s = ei[e];
        const long long d = ei[E + e];
        const float nrm = dinv[s] * dinv[d];
        const float4 hv = *(const float4*)(h + (size_t)s * 128 + q * 4);
        float* out = agg + (size_t)d * 128 + q * 4;
        atomicAdd(out + 0, hv.x * nrm);
        atomicAdd(out + 1, hv.y * nrm);
        atomicAdd(out + 2, hv.z * nrm);
        atomicAdd(out + 3, hv.w * nrm);
    }
}

// ---------------------------------------------------------------------------
// Fused: v = lrelu(agg[i,f] + h[i,f]*dinv[i]^2 + b[f]); pooled[g,f] = mean_i v
// One block per graph (128 nodes), thread f = feature. Coalesced on f.
// ---------------------------------------------------------------------------
__global__ __launch_bounds__(128) void pool_kernel(
    const float* __restrict__ agg, const float* __restrict__ h,
    const float* __restrict__ dinv, const float* __restrict__ b,
    float* __restrict__ pooled)
{
    const int g = blockIdx.x;
    const int f = threadIdx.x;
    const float bf = b[f];
    float sum = 0.0f;
    for (int j = 0; j < 128; ++j) {
        const int i = g * 128 + j;
        const float di = dinv[i];
        float v = agg[(size_t)i * 128 + f] + h[(size_t)i * 128 + f] * di * di + bf;
        v = (v >= 0.0f) ? v : NEG_SLOPE * v;
        sum += v;
    }
    pooled[(size_t)g * 128 + f] = sum * (1.0f / 128.0f);
}

// ---------------------------------------------------------------------------
extern "C" void kernel_launch(void* const* d_in, const int* in_sizes, int n_in,
                              void* d_out, int out_size, void* d_ws, size_t ws_size,
                              hipStream_t stream)
{
    (void)in_sizes; (void)n_in; (void)out_size; (void)ws_size;

    constexpr long long N = 131072, E = 2097152;
    constexpr int F = 128, H = 128, B = 1024, FC = 256;

    const float*     x   = (const float*)d_in[0];
    const long long* ei  = (const long long*)d_in[1];   // int64 [2,E]
    const float*     W   = (const float*)d_in[2];
    const float*     b   = (const float*)d_in[3];
    const float*     w1a = (const float*)d_in[4];
    const float*     b1a = (const float*)d_in[5];
    const float*     w1b = (const float*)d_in[6];
    const float*     b1b = (const float*)d_in[7];
    const float*     w2a = (const float*)d_in[8];
    const float*     b2a = (const float*)d_in[9];
    const float*     w2b = (const float*)d_in[10];
    const float*     b2b = (const float*)d_in[11];
    float* out = (float*)d_out;

    // Workspace layout (floats)
    float* ws     = (float*)d_ws;
    float* deg    = ws;                    // N
    float* dinv   = deg + N;               // N
    float* h      = dinv + N;              // N*H
    float* agg    = h + N * H;             // N*H
    float* pooled = agg + N * H;           // B*H
    float* t1     = pooled + (size_t)B * H;// B*FC
    float* t2     = t1 + (size_t)B * FC;   // B*FC

    // 0) zero deg + agg (atomic accumulation targets)
    zero_kernel<<<512, 256, 0, stream>>>(deg, (size_t)N);
    zero_kernel<<<8192, 256, 0, stream>>>(agg, (size_t)N * H);

    // 1) degrees + inverse-sqrt norm
    deg_kernel<<<4096, 256, 0, stream>>>(ei, deg, E);
    dinv_kernel<<<(int)(N / 256), 256, 0, stream>>>(deg, dinv, (int)N);

    // 2) h = x @ W   (WMMA f32 16x16x4), no bias/act
    wmma_gemm_f32<<<dim3(H / 16, (unsigned)(N / 128)), 256, 0, stream>>>(
        x, W, nullptr, h, (int)N, H, F, 0);

    // 3) edge gather/scatter (dominant: ~2 GB of L2/HBM traffic)
    edge_scatter<<<8192, 256, 0, stream>>>(h, ei, dinv, agg, E);

    // 4) fused self-loop + bias + lrelu + mean pool -> pooled [B,H]
    pool_kernel<<<B, 128, 0, stream>>>(agg, h, dinv, b, pooled);

    // 5) MLP heads (WMMA, bias + lrelu fused into first layer of each head)
    wmma_gemm_f32<<<dim3(FC / 16, B / 128), 256, 0, stream>>>(pooled, w1a, b1a, t1, B, FC, H, 1);
    wmma_gemm_f32<<<dim3(H  / 16, B / 128), 256, 0, stream>>>(t1,     w1b, b1b, out, B, H, FC, 0);
    wmma_gemm_f32<<<dim3(FC / 16, B / 128), 256, 0, stream>>>(pooled, w2a, b2a, t2, B, FC, H, 1);
    wmma_gemm_f32<<<dim3(H  / 16, B / 128), 256, 0, stream>>>(t2,     w2b, b2b, out + (size_t)B * H, B, H, FC, 0);
}